// External_Attention_83296595738811
// MI455X (gfx1250) — compile-verified
//
#include <hip/hip_runtime.h>
#include <hip/hip_bf16.h>
#include <math.h>

typedef __attribute__((ext_vector_type(2))) float v2f;
typedef __attribute__((ext_vector_type(8))) float v8f;
typedef __attribute__((ext_vector_type(4))) int   v4i;

#define KDIM    128
#define TILE_M  128
#define TILE_N  128
#define KCHUNK  32
#define NCHUNK  (KDIM / KCHUNK)
#define ASTRIDE 36      // 32 + 4 pad floats: 144B rows -> 16B aligned, bank-conflict free
#define CDIM    4096
#define HDIM    128

// ---- CDNA5 async global->LDS path (ASYNCcnt), with sync fallback --------------
#if defined(__gfx1250__) && __has_builtin(__builtin_amdgcn_global_load_async_to_lds_b128)
#define USE_ASYNC_LDS 1
#else
#define USE_ASYNC_LDS 0
#endif

#if USE_ASYNC_LDS
#define GLB_AS __attribute__((address_space(1)))
#define LDS_AS __attribute__((address_space(3)))

__device__ __forceinline__ void async_copy16(const void* g, void* l) {
    // global_load_async_to_lds_b128: 16B per lane, tracked by ASYNCcnt
    __builtin_amdgcn_global_load_async_to_lds_b128(
        (GLB_AS v4i*)g, (LDS_AS v4i*)l, 0, 0);
}
__device__ __forceinline__ void wait_async_le8() {
#if __has_builtin(__builtin_amdgcn_s_wait_asynccnt)
    __builtin_amdgcn_s_wait_asynccnt(8);
#else
    asm volatile("s_wait_asynccnt 0x8" ::: "memory");
#endif
}
__device__ __forceinline__ void wait_async_0() {
#if __has_builtin(__builtin_amdgcn_s_wait_asynccnt)
    __builtin_amdgcn_s_wait_asynccnt(0);
#else
    asm volatile("s_wait_asynccnt 0x0" ::: "memory");
#endif
}
#endif

// stage one 128x32 K-chunk of A and B into LDS (8 x b128 per lane-group pass)
__device__ __forceinline__ void stage_chunk(
    const float4* __restrict__ Ag, const float4* __restrict__ Bg,
    int m0, int n0, int kc, float* __restrict__ As, float* __restrict__ Bs, int tid)
{
    #pragma unroll
    for (int j = 0; j < 4; ++j) {
        int idx = tid + 256 * j;            // 0..1023
        int r   = idx >> 3;                 // 0..127
        int c4  = idx & 7;                  // 0..7 float4s within chunk
        const float4* ga = &Ag[(size_t)(m0 + r) * 32 + (kc >> 2) + c4];
        const float4* gb = &Bg[(size_t)(n0 + r) * 32 + (kc >> 2) + c4];
        float* la = &As[r * ASTRIDE + c4 * 4];
        float* lb = &Bs[r * ASTRIDE + c4 * 4];
#if USE_ASYNC_LDS
        async_copy16(ga, la);
        async_copy16(gb, lb);
#else
        *(float4*)la = *ga;
        *(float4*)lb = *gb;
#endif
    }
}

// C[M,N] = act( (A[M,128] x B[N,128]^T) * scale + bias ),  all dims multiples of tiles.
__global__ __launch_bounds__(256) void gemm_nt_epi(
    const float* __restrict__ A, const float* __restrict__ B,
    const float* __restrict__ bias, float* __restrict__ C,
    int M, int N, float scale, int do_relu)
{
    __shared__ float As[2][TILE_M * ASTRIDE];   // double buffered: 2 x 18.4KB
    __shared__ float Bs[2][TILE_N * ASTRIDE];   // double buffered: 2 x 18.4KB

    const int tid  = threadIdx.x;
    const int lane = tid & 31;
    const int wave = tid >> 5;
    const int m0   = blockIdx.x * TILE_M;
    const int n0   = blockIdx.y * TILE_N;

    const int wm = wave >> 1;           // 0..3 : 32-row slab
    const int wn = wave & 1;            // 0..1 : 64-col slab
    const int lm = lane & 15;           // row/col within 16x16 fragment
    const int lh = (lane >> 4) << 1;    // 0 or 2 : K-half select (ISA 32-bit A layout)

    v8f acc[2][4] = {};

    const float4* Ag = (const float4*)A;
    const float4* Bg = (const float4*)B;

    stage_chunk(Ag, Bg, m0, n0, 0, As[0], Bs[0], tid);

    for (int i = 0; i < NCHUNK; ++i) {
        const int nxt = i + 1;
        if (nxt < NCHUNK)
            stage_chunk(Ag, Bg, m0, n0, nxt * KCHUNK, As[nxt & 1], Bs[nxt & 1], tid);
#if USE_ASYNC_LDS
        if (nxt < NCHUNK) wait_async_le8();   // chunk i's 8 async ops complete
        else              wait_async_0();
#endif
        __syncthreads();

        const float* Abase = &As[i & 1][(wm * 32 + lm) * ASTRIDE + lh];
        const float* Bbase = &Bs[i & 1][(wn * 64 + lm) * ASTRIDE + lh];

        #pragma unroll
        for (int kk = 0; kk < KCHUNK; kk += 4) {
            v2f a[2], b[4];
            #pragma unroll
            for (int mi = 0; mi < 2; ++mi)
                a[mi] = *(const v2f*)(Abase + mi * 16 * ASTRIDE + kk);
            #pragma unroll
            for (int ni = 0; ni < 4; ++ni)
                b[ni] = *(const v2f*)(Bbase + ni * 16 * ASTRIDE + kk);
            #pragma unroll
            for (int mi = 0; mi < 2; ++mi)
                #pragma unroll
                for (int ni = 0; ni < 4; ++ni)
                    acc[mi][ni] = __builtin_amdgcn_wmma_f32_16x16x4_f32(
                        false, a[mi], false, b[ni], (short)0, acc[mi][ni],
                        false, false);
        }
        __syncthreads();   // protects buffer reuse two iterations ahead
    }

    // ---- epilogue: scale, bias, relu; cols contiguous across lanes ----
    const int rowhi = (lane >= 16) ? 8 : 0;
    #pragma unroll
    for (int mi = 0; mi < 2; ++mi) {
        #pragma unroll
        for (int ni = 0; ni < 4; ++ni) {
            int col  = n0 + wn * 64 + ni * 16 + lm;
            float bv = bias ? bias[col] : 0.0f;
            #pragma unroll
            for (int r = 0; r < 8; ++r) {
                int row = m0 + wm * 32 + mi * 16 + r + rowhi;
                float v = acc[mi][ni][r] * scale + bv;
                if (do_relu) v = fmaxf(v, 0.0f);
                C[(size_t)row * N + col] = v;
            }
        }
    }
}

// Per-row (B rows): softmax over C=4096 in place, exact top-k (first-index tie
// break, matching jax.lax.top_k on raw scores), mean of gathered V rows.
__global__ __launch_bounds__(256) void softmax_topk_gather(
    float* __restrict__ U,            // [B,C] raw scores in, softmax out (in place)
    const float* __restrict__ V,      // [C,H]
    float* __restrict__ sel,          // [B,H] mean of top-k V rows
    float* __restrict__ closs,
    const int* __restrict__ topk_ptr)
{
    __shared__ __align__(16) float s[CDIM];
    __shared__ float rv[256];
    __shared__ int   ri[256];
    __shared__ float vacc[HDIM];
    __shared__ float sh_max, sh_inv;
    __shared__ int   sh_idx;

    const int row = blockIdx.x;
    const int tid = threadIdx.x;
    float* Urow = U + (size_t)row * CDIM;
    const float4* Urow4 = (const float4*)Urow;
    float4* s4 = (float4*)s;

    // load (b128) + local max
    float lmax = -INFINITY;
    #pragma unroll
    for (int i = tid; i < CDIM / 4; i += 256) {
        float4 x = Urow4[i];
        s4[i] = x;
        lmax = fmaxf(fmaxf(lmax, fmaxf(x.x, x.y)), fmaxf(x.z, x.w));
    }
    rv[tid] = lmax;
    __syncthreads();
    for (int off = 128; off > 0; off >>= 1) {
        if (tid < off) rv[tid] = fmaxf(rv[tid], rv[tid + off]);
        __syncthreads();
    }
    if (tid == 0) sh_max = rv[0];
    __syncthreads();
    const float rmax = sh_max;

    // sum of exp
    float lsum = 0.0f;
    for (int i = tid; i < CDIM; i += 256) lsum += __expf(s[i] - rmax);
    rv[tid] = lsum;
    __syncthreads();
    for (int off = 128; off > 0; off >>= 1) {
        if (tid < off) rv[tid] += rv[tid + off];
        __syncthreads();
    }
    if (tid == 0) sh_inv = 1.0f / rv[0];
    __syncthreads();
    const float rinv = sh_inv;

    // softmax write-back as b128 (raw copy stays in LDS)
    float4* Uout4 = (float4*)Urow;
    #pragma unroll
    for (int i = tid; i < CDIM / 4; i += 256) {
        float4 x = s4[i];
        float4 o;
        o.x = __expf(x.x - rmax) * rinv;
        o.y = __expf(x.y - rmax) * rinv;
        o.z = __expf(x.z - rmax) * rinv;
        o.w = __expf(x.w - rmax) * rinv;
        Uout4[i] = o;
    }

    // top-k: iterative argmax with smallest-index tie break, mask winner
    if (tid < HDIM) vacc[tid] = 0.0f;
    const int K = topk_ptr[0];
    __syncthreads();
    for (int t = 0; t < K; ++t) {
        float bv = -INFINITY; int bi = CDIM;
        for (int i = tid; i < CDIM; i += 256) {
            float x = s[i];
            if (x > bv || (x == bv && i < bi)) { bv = x; bi = i; }
        }
        rv[tid] = bv; ri[tid] = bi;
        __syncthreads();
        for (int off = 128; off > 0; off >>= 1) {
            if (tid < off) {
                float ov = rv[tid + off]; int oi = ri[tid + off];
                if (ov > rv[tid] || (ov == rv[tid] && oi < ri[tid])) {
                    rv[tid] = ov; ri[tid] = oi;
                }
            }
            __syncthreads();
        }
        if (tid == 0) { sh_idx = ri[0]; s[ri[0]] = -INFINITY; }
        __syncthreads();
        const int idx = sh_idx;
        if (tid < HDIM) vacc[tid] += V[(size_t)idx * HDIM + tid];
        __syncthreads();
    }
    if (tid < HDIM)
        sel[(size_t)row * HDIM + tid] = vacc[tid] / (float)K;
    if (row == 0 && tid == 0) *closs = 0.0f;
}

extern "C" void kernel_launch(void* const* d_in, const int* in_sizes, int n_in,
                              void* d_out, int out_size, void* d_ws, size_t ws_size,
                              hipStream_t stream) {
    (void)in_sizes; (void)n_in; (void)out_size; (void)ws_size;

    const float* z    = (const float*)d_in[0];
    const float* Wq   = (const float*)d_in[1];
    const float* bq   = (const float*)d_in[2];
    const float* Wk   = (const float*)d_in[3];
    const float* bk   = (const float*)d_in[4];
    const float* mem  = (const float*)d_in[5];   // [C,1,H] == [C,H]
    const float* Wo   = (const float*)d_in[6];
    const float* bo   = (const float*)d_in[7];
    const int*   topk = (const int*)d_in[8];

    const int B = 16384, C = 4096, H = 128;

    float* out   = (float*)d_out;                 // [B,H]
    float* attn  = out  + (size_t)B * H;          // [B,C]
    float* closs = attn + (size_t)B * C;          // scalar

    float* kbuf = (float*)d_ws;                   // [C,H]
    float* qbuf = kbuf + (size_t)C * H;           // [B,H]
    float* sbuf = qbuf + (size_t)B * H;           // [B,H]

    const float inv_scale = 0.08838834764831845f; // 128^-0.5

    // k = relu(mem @ Wk^T + bk)
    gemm_nt_epi<<<dim3(C / TILE_M, H / TILE_N), 256, 0, stream>>>(
        mem, Wk, bk, kbuf, C, H, 1.0f, 1);
    // q = relu(z @ Wq^T + bq)
    gemm_nt_epi<<<dim3(B / TILE_M, H / TILE_N), 256, 0, stream>>>(
        z, Wq, bq, qbuf, B, H, 1.0f, 1);
    // u = (q @ k^T) / SCALE  -> raw scores into attn region
    gemm_nt_epi<<<dim3(B / TILE_M, C / TILE_N), 256, 0, stream>>>(
        qbuf, kbuf, nullptr, attn, B, C, inv_scale, 0);
    // softmax in place + top-k gather-mean (v == k)
    softmax_topk_gather<<<dim3(B), 256, 0, stream>>>(attn, kbuf, sbuf, closs, topk);
    // out = relu(sel @ Wo^T + bo)
    gemm_nt_epi<<<dim3(B / TILE_M, H / TILE_N), 256, 0, stream>>>(
        sbuf, Wo, bo, out, B, H, 1.0f, 1);
}